// GATv2_73933567033779
// MI455X (gfx1250) — compile-verified
//
#include <hip/hip_runtime.h>
#include <hip/hip_bf16.h>
#include <cstdint>

#define NEG_SLOPE 0.2f
#define LDS_STRIDE 132   // 128 + 4 floats pad: row m starts 4 banks after row m-1

typedef float v2f __attribute__((ext_vector_type(2)));
typedef float v8f __attribute__((ext_vector_type(8)));

__device__ __forceinline__ void atomic_add_f32(float* p, float v) {
    __hip_atomic_fetch_add(p, v, __ATOMIC_RELAXED, __HIP_MEMORY_SCOPE_AGENT);
}
__device__ __forceinline__ void atomic_max_f32(float* p, float v) {
    __hip_atomic_fetch_max(p, v, __ATOMIC_RELAXED, __HIP_MEMORY_SCOPE_AGENT);
}
__device__ __forceinline__ float lrelu(float v) {
    return v > 0.f ? v : NEG_SLOPE * v;
}

// ---------------------------------------------------------------------------
// Init helpers
// ---------------------------------------------------------------------------
__global__ void fill_f32(float* __restrict__ p, int n, float v) {
    int i = blockIdx.x * blockDim.x + threadIdx.x;
    if (i < n) p[i] = v;
}

__global__ void init_bias(float* __restrict__ p, const float* __restrict__ b,
                          int n, int C) {
    int i = blockIdx.x * blockDim.x + threadIdx.x;
    if (i < n) p[i] = b[i % C];
}

// ---------------------------------------------------------------------------
// Dual GEMM, K=128, N_out=128:  XL = X*Wl, XR = X*Wr   (f32 WMMA 16x16x4)
// Block: 256 threads = 8 waves; wave w owns output col-tile w (cols 16w..16w+15).
// One 16-row node tile per block, staged in LDS (padded stride, bank-conflict-free).
// A-frag (16x4 f32): lane<16 -> M=lane, K = k+{0,1}; lane>=16 -> M=lane-16, K=k+{2,3}
// C/D (16x16 f32):  VGPR r, lanes 0-15 -> (M=r, N=lane); lanes 16-31 -> (M=8+r).
// ---------------------------------------------------------------------------
__global__ __launch_bounds__(256) void gemm_dual_128(
    const float* __restrict__ X, const float* __restrict__ Wl,
    const float* __restrict__ Wr, float* __restrict__ XL,
    float* __restrict__ XR, int N) {
    __shared__ float tile[16 * LDS_STRIDE];
    const int row0 = blockIdx.x * 16;
    const int tid  = threadIdx.x;
    for (int i = tid; i < 16 * 128; i += 256) {
        int r = i >> 7, c = i & 127;
        int row = row0 + r;
        tile[r * LDS_STRIDE + c] = (row < N) ? X[(size_t)row * 128 + c] : 0.f;
    }
    __syncthreads();

    const int wave  = tid >> 5;
    const int lane  = tid & 31;
    const int m     = lane & 15;
    const int khalf = (lane >> 4) * 2;          // 0 or 2
    const int ncol  = wave * 16 + m;            // this lane's B/C column

    v8f accL = {};
    v8f accR = {};
#pragma unroll
    for (int k = 0; k < 128; k += 4) {
        v2f a, bl, br;
        a.x  = tile[m * LDS_STRIDE + k + khalf];
        a.y  = tile[m * LDS_STRIDE + k + khalf + 1];
        bl.x = Wl[(k + khalf) * 128 + ncol];
        bl.y = Wl[(k + khalf + 1) * 128 + ncol];
        br.x = Wr[(k + khalf) * 128 + ncol];
        br.y = Wr[(k + khalf + 1) * 128 + ncol];
        accL = __builtin_amdgcn_wmma_f32_16x16x4_f32(false, a, false, bl,
                                                     (short)0, accL, false, false);
        accR = __builtin_amdgcn_wmma_f32_16x16x4_f32(false, a, false, br,
                                                     (short)0, accR, false, false);
    }

    const int rbase = row0 + (lane >> 4) * 8;
#pragma unroll
    for (int r = 0; r < 8; ++r) {
        int row = rbase + r;
        if (row < N) {
            XL[(size_t)row * 128 + ncol] = accL[r];
            XR[(size_t)row * 128 + ncol] = accR[r];
        }
    }
}

// ---------------------------------------------------------------------------
// Dual GEMM, K=128, N_out=40 (padded to 48 = 3 col tiles), ReLU on the A load.
// Block: 96 threads = 3 waves, wave w -> col tile w, for both Wl and Wr.
// ---------------------------------------------------------------------------
__global__ __launch_bounds__(96) void gemm_dual_40(
    const float* __restrict__ H, const float* __restrict__ Wl,
    const float* __restrict__ Wr, float* __restrict__ HL,
    float* __restrict__ HR, int N) {
    __shared__ float tile[16 * LDS_STRIDE];
    const int row0 = blockIdx.x * 16;
    const int tid  = threadIdx.x;
    for (int i = tid; i < 16 * 128; i += 96) {
        int r = i >> 7, c = i & 127;
        int row = row0 + r;
        tile[r * LDS_STRIDE + c] =
            (row < N) ? fmaxf(H[(size_t)row * 128 + c], 0.f) : 0.f;  // fused ReLU
    }
    __syncthreads();

    const int wave   = tid >> 5;
    const int lane   = tid & 31;
    const int m      = lane & 15;
    const int khalf  = (lane >> 4) * 2;
    const int col    = wave * 16 + m;
    const bool valid = col < 40;

    v8f accL = {};
    v8f accR = {};
#pragma unroll
    for (int k = 0; k < 128; k += 4) {
        v2f a, bl, br;
        a.x  = tile[m * LDS_STRIDE + k + khalf];
        a.y  = tile[m * LDS_STRIDE + k + khalf + 1];
        bl.x = valid ? Wl[(k + khalf) * 40 + col] : 0.f;
        bl.y = valid ? Wl[(k + khalf + 1) * 40 + col] : 0.f;
        br.x = valid ? Wr[(k + khalf) * 40 + col] : 0.f;
        br.y = valid ? Wr[(k + khalf + 1) * 40 + col] : 0.f;
        accL = __builtin_amdgcn_wmma_f32_16x16x4_f32(false, a, false, bl,
                                                     (short)0, accL, false, false);
        accR = __builtin_amdgcn_wmma_f32_16x16x4_f32(false, a, false, br,
                                                     (short)0, accR, false, false);
    }

    const int rbase = row0 + (lane >> 4) * 8;
#pragma unroll
    for (int r = 0; r < 8; ++r) {
        int row = rbase + r;
        if (row < N && valid) {
            HL[(size_t)row * 40 + col] = accL[r];
            HR[(size_t)row * 40 + col] = accR[r];
        }
    }
}

// ---------------------------------------------------------------------------
// Layer-1 edge scoring: thread per (edge, head). H=8, C=16.
// score = sum_c att[h][c] * leaky_relu(xl[src][h][c] + xr[dst][h][c])
// + atomic f32 max into per-(dst,head) running max for the edge softmax.
// ---------------------------------------------------------------------------
__global__ void edge_score_h8(const long long* __restrict__ ei, int E, int Ep,
                              const float* __restrict__ XL,
                              const float* __restrict__ XR,
                              const float* __restrict__ att,
                              float* __restrict__ score,
                              float* __restrict__ smax) {
    unsigned t = blockIdx.x * blockDim.x + threadIdx.x;
    if (t >= (unsigned)Ep * 8u) return;
    unsigned e = t >> 3;
    int h = (int)(t & 7u);
    int s, d;
    if (e < (unsigned)E) {
        s = (int)ei[e];
        d = (int)ei[(long long)E + e];
    } else {
        s = d = (int)(e - (unsigned)E);  // self loop
    }
    const float4* a4 = (const float4*)(XL + (size_t)s * 128 + h * 16);
    const float4* b4 = (const float4*)(XR + (size_t)d * 128 + h * 16);
    const float4* w4 = (const float4*)(att + h * 16);
    float acc = 0.f;
#pragma unroll
    for (int i = 0; i < 4; ++i) {
        float4 a = a4[i], b = b4[i], w = w4[i];
        acc += lrelu(a.x + b.x) * w.x + lrelu(a.y + b.y) * w.y +
               lrelu(a.z + b.z) * w.z + lrelu(a.w + b.w) * w.w;
    }
    score[t] = acc;
    atomic_max_f32(&smax[(size_t)d * 8 + h], acc);
}

// ---------------------------------------------------------------------------
// Layer-2 edge scoring: thread per edge. H=1, C=40.
// ---------------------------------------------------------------------------
__global__ void edge_score_h1(const long long* __restrict__ ei, int E, int Ep,
                              const float* __restrict__ HL,
                              const float* __restrict__ HR,
                              const float* __restrict__ att,
                              float* __restrict__ score,
                              float* __restrict__ smax) {
    int e = blockIdx.x * blockDim.x + threadIdx.x;
    if (e >= Ep) return;
    int s, d;
    if (e < E) {
        s = (int)ei[e];
        d = (int)ei[(long long)E + e];
    } else {
        s = d = e - E;
    }
    const float4* a4 = (const float4*)(HL + (size_t)s * 40);
    const float4* b4 = (const float4*)(HR + (size_t)d * 40);
    const float4* w4 = (const float4*)att;
    float acc = 0.f;
#pragma unroll
    for (int i = 0; i < 10; ++i) {
        float4 a = a4[i], b = b4[i], w = w4[i];
        acc += lrelu(a.x + b.x) * w.x + lrelu(a.y + b.y) * w.y +
               lrelu(a.z + b.z) * w.z + lrelu(a.w + b.w) * w.w;
    }
    score[e] = acc;
    atomic_max_f32(&smax[d], acc);
}

// ---------------------------------------------------------------------------
// exp(score - max) and denominator accumulation. In-place on score.
// ---------------------------------------------------------------------------
template <int H>
__global__ void edge_exp(const long long* __restrict__ ei, int E, int Ep,
                         float* __restrict__ score,
                         const float* __restrict__ smax,
                         float* __restrict__ denom) {
    unsigned t = blockIdx.x * blockDim.x + threadIdx.x;
    if (t >= (unsigned)Ep * (unsigned)H) return;
    unsigned e = t / (unsigned)H;
    int h = (int)(t % (unsigned)H);
    int d = (e < (unsigned)E) ? (int)ei[(long long)E + e]
                              : (int)(e - (unsigned)E);
    float v = __expf(score[t] - smax[(size_t)d * H + h]);
    score[t] = v;
    atomic_add_f32(&denom[(size_t)d * H + h], v);
}

// ---------------------------------------------------------------------------
// Aggregation: thread per (edge, channel).  out[dst][c] += xl[src][c]*alpha.
// ---------------------------------------------------------------------------
template <int C, int CH, int H>
__global__ void edge_aggr(const long long* __restrict__ ei, int E, int Ep,
                          const float* __restrict__ SRC,
                          const float* __restrict__ expd,
                          const float* __restrict__ denom,
                          float* __restrict__ out) {
    unsigned t = blockIdx.x * blockDim.x + threadIdx.x;
    if (t >= (unsigned)Ep * (unsigned)C) return;
    unsigned e = t / (unsigned)C;
    int c = (int)(t % (unsigned)C);
    int h = c / CH;
    int s, d;
    if (e < (unsigned)E) {
        s = (int)ei[e];
        d = (int)ei[(long long)E + e];
    } else {
        s = d = (int)(e - (unsigned)E);
    }
    float alpha = expd[(size_t)e * H + h] / denom[(size_t)d * H + h];
    atomic_add_f32(&out[(size_t)d * C + c], SRC[(size_t)s * C + c] * alpha);
}

// ---------------------------------------------------------------------------
// Host-side orchestration
// ---------------------------------------------------------------------------
extern "C" void kernel_launch(void* const* d_in, const int* in_sizes, int n_in,
                              void* d_out, int out_size, void* d_ws,
                              size_t ws_size, hipStream_t stream) {
    (void)n_in; (void)out_size; (void)ws_size;
    const float*     x    = (const float*)d_in[0];
    const long long* ei   = (const long long*)d_in[1];
    const float*     W1l  = (const float*)d_in[2];
    const float*     W1r  = (const float*)d_in[3];
    const float*     att1 = (const float*)d_in[4];
    const float*     b1   = (const float*)d_in[5];
    const float*     W2l  = (const float*)d_in[6];
    const float*     W2r  = (const float*)d_in[7];
    const float*     att2 = (const float*)d_in[8];
    const float*     b2   = (const float*)d_in[9];

    const int N  = in_sizes[0] / 128;
    const int E  = in_sizes[1] / 2;
    const int Ep = E + N;   // with self loops

    // Workspace arena (everything fits comfortably in L2-backed HBM; the hot
    // gather/scatter set — xl, xr, h1, hl2, hr2 — is < 192 MB -> L2 resident).
    char*  ws  = (char*)d_ws;
    size_t off = 0;
    auto arena = [&](size_t bytes) -> float* {
        float* p = (float*)(ws + off);
        off += (bytes + 255) & ~(size_t)255;
        return p;
    };
    float* XL   = arena((size_t)N * 128 * 4);
    float* XR   = arena((size_t)N * 128 * 4);
    float* H1   = arena((size_t)N * 128 * 4);
    float* SC   = arena((size_t)Ep * 8 * 4);   // scores/expd (layer2 reuses prefix)
    float* SMAX = arena((size_t)N * 8 * 4);
    float* DEN  = arena((size_t)N * 8 * 4);
    float* HL   = XL;   // layer-2 reuses (N*40 <= N*128), XL dead by then
    float* HR   = XR;

    const int rowTiles = (N + 15) / 16;
    auto blocks = [](unsigned total, unsigned bs) { return (total + bs - 1) / bs; };

    // ---- Layer 1 ----
    fill_f32<<<blocks(N * 8, 256), 256, 0, stream>>>(SMAX, N * 8, -INFINITY);
    fill_f32<<<blocks(N * 8, 256), 256, 0, stream>>>(DEN, N * 8, 0.f);
    init_bias<<<blocks(N * 128, 256), 256, 0, stream>>>(H1, b1, N * 128, 128);

    gemm_dual_128<<<rowTiles, 256, 0, stream>>>(x, W1l, W1r, XL, XR, N);

    edge_score_h8<<<blocks((unsigned)Ep * 8u, 256), 256, 0, stream>>>(
        ei, E, Ep, XL, XR, att1, SC, SMAX);
    edge_exp<8><<<blocks((unsigned)Ep * 8u, 256), 256, 0, stream>>>(
        ei, E, Ep, SC, SMAX, DEN);
    edge_aggr<128, 16, 8><<<blocks((unsigned)Ep * 128u, 256), 256, 0, stream>>>(
        ei, E, Ep, XL, SC, DEN, H1);

    // ---- Layer 2 (ReLU fused into GEMM-2 A-tile load) ----
    fill_f32<<<blocks(N, 256), 256, 0, stream>>>(SMAX, N, -INFINITY);
    fill_f32<<<blocks(N, 256), 256, 0, stream>>>(DEN, N, 0.f);

    gemm_dual_40<<<rowTiles, 96, 0, stream>>>(H1, W2l, W2r, HL, HR, N);

    edge_score_h1<<<blocks(Ep, 256), 256, 0, stream>>>(ei, E, Ep, HL, HR, att2,
                                                       SC, SMAX);
    edge_exp<1><<<blocks(Ep, 256), 256, 0, stream>>>(ei, E, Ep, SC, SMAX, DEN);

    float* out = (float*)d_out;
    init_bias<<<blocks(N * 40, 256), 256, 0, stream>>>(out, b2, N * 40, 40);
    edge_aggr<40, 40, 1><<<blocks((unsigned)Ep * 40u, 256), 256, 0, stream>>>(
        ei, E, Ep, HL, SC, DEN, out);
}